// STRCTCLoss_36043365548613
// MI455X (gfx1250) — compile-verified
//
#include <hip/hip_runtime.h>
#include <math.h>

// CTC loss: B=256, T=256, C=1000, S=32, L=2S+1=65, blank=0.
// d_in[0]: preds  float32 [B*T*C]
// d_in[1]: targets int     [B*S]   (labels in [1,C))
// d_out  : float32 [1]  mean(loss_b / S)
// d_ws   : >= B floats scratch for per-batch losses.

#define CTC_B 256
#define CTC_T 256
#define CTC_C 1000
#define CTC_S 32
#define CTC_L 65
#define NEG_INF (-1.0e30f)
#define TSTR 132          // padded LDS tile row stride (floats); 132*4=528B, 16B aligned
#define CHUNK 128         // columns staged per LDS tile pass

typedef float v2f __attribute__((ext_vector_type(2)));
typedef float v8f __attribute__((ext_vector_type(8)));

__device__ __forceinline__ float lae(float x, float y) {
    // logaddexp with NEG_INF sentinel (finite -1e30, so no NaN from inf-inf)
    float m = fmaxf(x, y);
    float d = fminf(x, y) - m;
    return m + log1pf(__expf(d));
}

__global__ __launch_bounds__(256)
void ctc_batch_kernel(const float* __restrict__ preds,
                      const int* __restrict__ targets,
                      float* __restrict__ ws) {
    __shared__ float lp[CTC_T * 33];      // gathered log-probs: [t][slot], slot0=blank
    __shared__ float tile[16 * TSTR];     // 16 t-rows x CHUNK c-cols staging
    __shared__ float rowmax_s[16];
    __shared__ float partials[8 * 16];    // per-wave WMMA row sums
    __shared__ float lse_s[16];
    __shared__ int   cls_s[33];           // slot -> class (0 = blank)

    const int b    = blockIdx.x;
    const int tid  = threadIdx.x;
    // wave id: force scalar so loops below are uniform (SGPR bounds, EXEC stays all-1s
    // across the WMMA loop, no per-iteration exec masking)
    const int w    = __builtin_amdgcn_readfirstlane(tid >> 5);
    const int lane = tid & 31;
    const int m16  = lane & 15;           // WMMA row index for this lane
    const int cOff = (lane >> 4) << 1;    // 0 for lanes 0-15, 2 for lanes 16-31

    if (tid < 33)
        cls_s[tid] = (tid == 0) ? 0 : targets[b * CTC_S + (tid - 1)];
    __syncthreads();

    // ---------------- Phase 1: lse + gather into lp[][] ----------------
    for (int tt = 0; tt < 16; ++tt) {
        const int t0 = tt * 16;

        // rowmax: wave w handles rows w and w+8; float4 streaming (coalesced 512B/iter)
        for (int rr = 0; rr < 2; ++rr) {
            const int r = w + rr * 8;
            const float4* rowp4 = reinterpret_cast<const float4*>(
                preds + (size_t)(b * CTC_T + t0 + r) * CTC_C);
            float mx = -3.0e38f;
            for (int c4 = lane; c4 < (CTC_C / 4); c4 += 32) {   // 250 float4s
                float4 v = rowp4[c4];
                mx = fmaxf(mx, fmaxf(fmaxf(v.x, v.y), fmaxf(v.z, v.w)));
            }
            for (int off = 16; off; off >>= 1) mx = fmaxf(mx, __shfl_xor(mx, off));
            if (lane == 0) rowmax_s[r] = mx;
        }
        __syncthreads();

        const float rmax = rowmax_s[m16];
        v8f acc = {0.f, 0.f, 0.f, 0.f, 0.f, 0.f, 0.f, 0.f};
        v2f bones; bones.x = 1.0f; bones.y = 1.0f;   // B = ones(4x16)

        for (int c0 = 0; c0 < CTC_C; c0 += CHUNK) {
            const int Wact = (CTC_C - c0 < CHUNK) ? (CTC_C - c0) : CHUNK;  // 128 or 104
            const int W4   = Wact >> 2;
            __syncthreads();  // protect tile from previous chunk's readers
            // stage 16 x Wact as float4: global_load_b128 -> ds_store_b128
            for (int i = tid; i < 16 * (CHUNK / 4); i += 256) {
                const int r  = i >> 5;           // 32 float4s per row slot
                const int c4 = i & 31;
                if (c4 < W4) {
                    float4 v = *reinterpret_cast<const float4*>(
                        preds + (size_t)(b * CTC_T + t0 + r) * CTC_C + c0 + (c4 << 2));
                    *reinterpret_cast<float4*>(&tile[r * TSTR + (c4 << 2)]) = v;
                }
            }
            __syncthreads();
            // sum-of-exp via V_WMMA_F32_16X16X4_F32: D[m][*] += sum_k exp(x[m][k])
            const int nch = W4;                   // 4-wide K chunks
            for (int q = w; q < nch; q += 8) {
                const float* src = &tile[m16 * TSTR + (q << 2) + cOff];
                v2f a;
                a.x = __expf(src[0] - rmax);
                a.y = __expf(src[1] - rmax);
                acc = __builtin_amdgcn_wmma_f32_16x16x4_f32(
                    false, a, false, bones, (short)0, acc, false, false);
            }
        }
        __syncthreads();
        // column 0 of D lives in lanes 0 (rows 0..7) and 16 (rows 8..15)
        if (lane == 0) {
            #pragma unroll
            for (int j = 0; j < 8; ++j) partials[w * 16 + j] = acc[j];
        }
        if (lane == 16) {
            #pragma unroll
            for (int j = 0; j < 8; ++j) partials[w * 16 + 8 + j] = acc[j];
        }
        __syncthreads();
        if (tid < 16) {
            float s = 0.f;
            #pragma unroll
            for (int ww = 0; ww < 8; ++ww) s += partials[ww * 16 + tid];
            lse_s[tid] = rowmax_s[tid] + logf(s);
        }
        __syncthreads();
        // gather 16 rows x 33 slots (blank + 32 label classes); L2 hits
        for (int e = tid; e < 16 * 33; e += 256) {
            const int r = e / 33;
            const int j = e - r * 33;
            const int cls = cls_s[j];
            const float v =
                preds[(size_t)(b * CTC_T + t0 + r) * CTC_C + cls] - lse_s[r];
            lp[(t0 + r) * 33 + j] = v;
        }
        __syncthreads();
    }

    // ---------------- Phase 2: alpha recursion (wave 0, registers only) ----
    if (tid < 32) {
        const int l = lane;
        const int sA = l;          // state l        (0..31)
        const int sB = l + 32;     // state l+32     (32..63)
        const int slotA = (sA & 1) ? (1 + (sA >> 1)) : 0;
        const int slotB = (sB & 1) ? (1 + (sB >> 1)) : 0;

        bool allowA = false, allowB = false;
        if ((sA & 1) && sA >= 3) {
            const int j = sA >> 1;                 // j >= 1
            allowA = (cls_s[j + 1] != cls_s[j]);   // targets[j] != targets[j-1]
        }
        if (sB & 1) {                              // sB >= 33 so s>=3 holds
            const int j = sB >> 1;
            allowB = (cls_s[j + 1] != cls_s[j]);
        }

        // t = 0 init: only states 0 and 1 reachable
        float aA = (l < 2) ? lp[slotA] : NEG_INF;
        float aB = NEG_INF;
        float aC = NEG_INF;        // state 64 (blank), meaningful on lane 0

        for (int t = 1; t < CTC_T; ++t) {
            const float u1A = __shfl_up(aA, 1);
            const float u1B = __shfl_up(aB, 1);
            const float u2A = __shfl_up(aA, 2);
            const float u2B = __shfl_up(aB, 2);
            const float a30 = __shfl(aA, 30);
            const float a31 = __shfl(aA, 31);
            const float b31 = __shfl(aB, 31);

            const float sh1A = (l >= 1) ? u1A : NEG_INF;
            const float sh2A = (l >= 2) ? u2A : NEG_INF;
            const float sh1B = (l >= 1) ? u1B : a31;                 // alpha[31]
            const float sh2B = (l >= 2) ? u2B : ((l == 1) ? a31 : a30);
            const float sh1C = b31;                                  // alpha[63]

            const float* lprow = lp + t * 33;
            float mA = lae(aA, sh1A);
            if (allowA) mA = lae(mA, sh2A);
            float mB = lae(aB, sh1B);
            if (allowB) mB = lae(mB, sh2B);
            float mC = lae(aC, sh1C);

            aA = mA + lprow[slotA];
            aB = mB + lprow[slotB];
            aC = mC + lprow[0];
        }
        const float alpha63 = __shfl(aB, 31);
        if (l == 0) {
            const float ll = lae(aC, alpha63);   // logaddexp(alpha[64], alpha[63])
            float loss = -ll;
            if (loss >= 1e10f) loss = 0.0f;      // zero_infinity
            ws[b] = loss;
        }
    }
}

__global__ void ctc_final_kernel(const float* __restrict__ ws,
                                 float* __restrict__ out) {
    if (blockIdx.x == 0 && threadIdx.x == 0) {
        float s = 0.0f;
        for (int i = 0; i < CTC_B; ++i) s += ws[i];   // fixed order: deterministic
        out[0] = s / ((float)CTC_B * (float)CTC_S);
    }
}

extern "C" void kernel_launch(void* const* d_in, const int* in_sizes, int n_in,
                              void* d_out, int out_size, void* d_ws, size_t ws_size,
                              hipStream_t stream) {
    const float* preds  = (const float*)d_in[0];
    const int*   tgts   = (const int*)d_in[1];
    float*       out    = (float*)d_out;
    float*       wsbuf  = (float*)d_ws;

    ctc_batch_kernel<<<CTC_B, 256, 0, stream>>>(preds, tgts, wsbuf);
    ctc_final_kernel<<<1, 32, 0, stream>>>(wsbuf, out);
}